// OmniAnomaly_89163521065570
// MI455X (gfx1250) — compile-verified
//
#include <hip/hip_runtime.h>
#include <math.h>

// OmniAnomaly fused forward: GRU0 -> GRU1 -> encoder -> reparam -> decoder.
// Single 256-thread (8-wave) workgroup; all mat-vecs via V_WMMA_F32_16X16X4_F32
// (exact f32 math). Latency-bound problem: weights prefetched up front.

typedef __attribute__((ext_vector_type(2))) float v2f;
typedef __attribute__((ext_vector_type(8))) float v8f;

#define NT 256
#define NW 8  // waves per block (wave32)

struct Params {
    const float *x, *hidden, *eps;
    const float *w_ih0, *w_hh0, *b_ih0, *b_hh0;
    const float *w_ih1, *w_hh1, *b_ih1, *b_hh1;
    const float *enc_w1, *enc_b1, *enc_a1;
    const float *enc_w2, *enc_b2, *enc_a2;
    const float *enc_w3, *enc_b3;
    const float *dec_w1, *dec_b1, *dec_a1;
    const float *dec_w2, *dec_b2, *dec_a2;
    const float *dec_w3, *dec_b3;
    float *out;
};

__device__ __forceinline__ float sigmoidf_(float v) { return 1.0f / (1.0f + expf(-v)); }

// Compute out[row0 .. row0+15] = W[row0:row0+16, :] @ xv  (W row-major, stride K)
// using chained V_WMMA_F32_16X16X4_F32. xv broadcast into all 16 B columns so
// every column of D carries the mat-vec; lanes 0 and 16 write rows 0-7 / 8-15.
// Must be called with all 32 lanes of the wave active (wave-uniform branch).
__device__ __forceinline__ void matvec16_wmma(const float* __restrict__ W, int K,
                                              const float* __restrict__ xv,
                                              float* __restrict__ out,
                                              int row0, int lane) {
    v8f acc = {};
    const int m  = lane & 15;            // M row within tile (both half-waves)
    const int kh = (lane >> 4) << 1;     // K sub-offset: 0 for lanes 0-15, 2 for 16-31
    const float* wrow = W + (row0 + m) * K + kh;
#pragma unroll 4
    for (int k = 0; k < K; k += 4) {
        v2f a, b;
        a.x = wrow[k];
        a.y = wrow[k + 1];
        b.x = xv[k + kh];
        b.y = xv[k + kh + 1];
        // (neg_a, A, neg_b, B, c_mod, C, reuse_a, reuse_b)
        acc = __builtin_amdgcn_wmma_f32_16x16x4_f32(false, a, false, b,
                                                    (short)0, acc, false, false);
    }
    if (m == 0) {
        const int base = row0 + ((lane >> 4) << 3);
#pragma unroll
        for (int v = 0; v < 8; ++v) out[base + v] = acc[v];
    }
}

__global__ __launch_bounds__(NT) void omnianomaly_fused_kernel(Params p) {
    const int tid  = threadIdx.x;
    const int lane = tid & 31;
    const int wid  = tid >> 5;

    __shared__ float x_s[128];
    __shared__ float h0p[32], h1p[32];     // previous hidden states
    __shared__ float h0n[32], h1n[32];     // new hidden states
    __shared__ float gi[96], gh[96];       // GRU gate pre-activations
    __shared__ float vecA[32], vecB[32];   // MLP ping-pong vectors
    __shared__ float raw[128];             // raw mat-vec outputs
    __shared__ float encv[16];             // mu (0-7) | logvar (8-15)
    __shared__ float zz[8];                // latent sample

    // --- Prefetch later-stage weights (one lane per 128B line) so HBM latency
    // overlaps the GRU0 dependency chain. Emits global_prefetch_b8.
    {
        const int CL = 32;  // floats per 128B cacheline
        for (int i = tid * CL; i < 96 * 32; i += NT * CL) {
            __builtin_prefetch(p.w_ih1 + i, 0, 0);
            __builtin_prefetch(p.w_hh1 + i, 0, 0);
        }
        for (int i = tid * CL; i < 128 * 32; i += NT * CL)
            __builtin_prefetch(p.dec_w3 + i, 0, 0);
    }

    // --- Stage in inputs
    if (tid < 128) x_s[tid] = p.x[tid];
    if (tid < 32) { h0p[tid] = p.hidden[tid]; h1p[tid] = p.hidden[32 + tid]; }
    __syncthreads();

    // ================= GRU cell 0 =================
    // gi = w_ih0 @ x (96x128, 6 tiles)  |  gh = w_hh0 @ h0p (96x32, 6 tiles)
    for (int t = wid; t < 12; t += NW) {
        if (t < 6) matvec16_wmma(p.w_ih0, 128, x_s, gi, t * 16, lane);
        else       matvec16_wmma(p.w_hh0, 32, h0p, gh, (t - 6) * 16, lane);
    }
    __syncthreads();
    if (tid < 32) {
        float ir = gi[tid]      + p.b_ih0[tid],      hr = gh[tid]      + p.b_hh0[tid];
        float iz = gi[32 + tid] + p.b_ih0[32 + tid], hz = gh[32 + tid] + p.b_hh0[32 + tid];
        float in_ = gi[64 + tid] + p.b_ih0[64 + tid], hn = gh[64 + tid] + p.b_hh0[64 + tid];
        float r = sigmoidf_(ir + hr);
        float z = sigmoidf_(iz + hz);
        float n = tanhf(in_ + r * hn);
        float h = (1.0f - z) * n + z * h0p[tid];
        h0n[tid] = h;
        p.out[144 + tid] = h;          // hidden_new[0]
    }
    __syncthreads();

    // ================= GRU cell 1 =================
    for (int t = wid; t < 12; t += NW) {
        if (t < 6) matvec16_wmma(p.w_ih1, 32, h0n, gi, t * 16, lane);
        else       matvec16_wmma(p.w_hh1, 32, h1p, gh, (t - 6) * 16, lane);
    }
    __syncthreads();
    if (tid < 32) {
        float ir = gi[tid]      + p.b_ih1[tid],      hr = gh[tid]      + p.b_hh1[tid];
        float iz = gi[32 + tid] + p.b_ih1[32 + tid], hz = gh[32 + tid] + p.b_hh1[32 + tid];
        float in_ = gi[64 + tid] + p.b_ih1[64 + tid], hn = gh[64 + tid] + p.b_hh1[64 + tid];
        float r = sigmoidf_(ir + hr);
        float z = sigmoidf_(iz + hz);
        float n = tanhf(in_ + r * hn);
        float h = (1.0f - z) * n + z * h1p[tid];
        h1n[tid] = h;
        p.out[176 + tid] = h;          // hidden_new[1]
    }
    __syncthreads();

    // ================= Encoder =================
    for (int t = wid; t < 2; t += NW) matvec16_wmma(p.enc_w1, 32, h1n, raw, t * 16, lane);
    __syncthreads();
    if (tid < 32) {
        float v = raw[tid] + p.enc_b1[tid];
        float a = p.enc_a1[0];
        vecA[tid] = (v >= 0.0f) ? v : a * v;
    }
    __syncthreads();

    for (int t = wid; t < 2; t += NW) matvec16_wmma(p.enc_w2, 32, vecA, raw, t * 16, lane);
    __syncthreads();
    if (tid < 32) {
        float v = raw[tid] + p.enc_b2[tid];
        float a = p.enc_a2[0];
        vecB[tid] = (v >= 0.0f) ? v : a * v;
    }
    __syncthreads();

    if (wid == 0) matvec16_wmma(p.enc_w3, 32, vecB, raw, 0, lane);  // 16 rows
    __syncthreads();
    if (tid < 16) {
        float v = raw[tid] + p.enc_b3[tid];
        encv[tid] = v;
        p.out[128 + tid] = v;          // mu (0-7), logvar (8-15)
    }
    __syncthreads();

    // ================= Reparameterize =================
    if (tid < 8) {
        float mu = encv[tid], lv = encv[8 + tid];
        zz[tid] = mu + p.eps[tid] * expf(0.5f * lv);
    }
    __syncthreads();

    // ================= Decoder =================
    for (int t = wid; t < 2; t += NW) matvec16_wmma(p.dec_w1, 8, zz, raw, t * 16, lane);
    __syncthreads();
    if (tid < 32) {
        float v = raw[tid] + p.dec_b1[tid];
        float a = p.dec_a1[0];
        vecA[tid] = (v >= 0.0f) ? v : a * v;
    }
    __syncthreads();

    for (int t = wid; t < 2; t += NW) matvec16_wmma(p.dec_w2, 32, vecA, raw, t * 16, lane);
    __syncthreads();
    if (tid < 32) {
        float v = raw[tid] + p.dec_b2[tid];
        float a = p.dec_a2[0];
        vecB[tid] = (v >= 0.0f) ? v : a * v;
    }
    __syncthreads();

    for (int t = wid; t < 8; t += NW) matvec16_wmma(p.dec_w3, 32, vecB, raw, t * 16, lane);
    __syncthreads();
    if (tid < 128) {
        float v = raw[tid] + p.dec_b3[tid];
        p.out[tid] = sigmoidf_(v);     // decoded
    }
}

extern "C" void kernel_launch(void* const* d_in, const int* in_sizes, int n_in,
                              void* d_out, int out_size, void* d_ws, size_t ws_size,
                              hipStream_t stream) {
    (void)in_sizes; (void)n_in; (void)out_size; (void)d_ws; (void)ws_size;
    Params p;
    p.x      = (const float*)d_in[0];
    p.hidden = (const float*)d_in[1];
    p.eps    = (const float*)d_in[2];
    p.w_ih0  = (const float*)d_in[3];
    p.w_hh0  = (const float*)d_in[4];
    p.b_ih0  = (const float*)d_in[5];
    p.b_hh0  = (const float*)d_in[6];
    p.w_ih1  = (const float*)d_in[7];
    p.w_hh1  = (const float*)d_in[8];
    p.b_ih1  = (const float*)d_in[9];
    p.b_hh1  = (const float*)d_in[10];
    p.enc_w1 = (const float*)d_in[11];
    p.enc_b1 = (const float*)d_in[12];
    p.enc_a1 = (const float*)d_in[13];
    p.enc_w2 = (const float*)d_in[14];
    p.enc_b2 = (const float*)d_in[15];
    p.enc_a2 = (const float*)d_in[16];
    p.enc_w3 = (const float*)d_in[17];
    p.enc_b3 = (const float*)d_in[18];
    p.dec_w1 = (const float*)d_in[19];
    p.dec_b1 = (const float*)d_in[20];
    p.dec_a1 = (const float*)d_in[21];
    p.dec_w2 = (const float*)d_in[22];
    p.dec_b2 = (const float*)d_in[23];
    p.dec_a2 = (const float*)d_in[24];
    p.dec_w3 = (const float*)d_in[25];
    p.dec_b3 = (const float*)d_in[26];
    p.out    = (float*)d_out;

    omnianomaly_fused_kernel<<<1, NT, 0, stream>>>(p);
}